// GCNEncoder_60876866453592
// MI455X (gfx1250) — compile-verified
//
#include <hip/hip_runtime.h>

#define NN 50000
#define NE 800000
#define CH 128

typedef float v2f __attribute__((ext_vector_type(2)));
typedef float v8f __attribute__((ext_vector_type(8)));

// ---------------- degree / norm ----------------

__global__ void init_deg_kernel(float* __restrict__ deg) {
  int i = blockIdx.x * blockDim.x + threadIdx.x;
  if (i < NN) deg[i] = 1.0f;  // self-loop contributes 1 to every node's degree
}

__global__ void deg_acc_kernel(const long long* __restrict__ ei, float* __restrict__ deg) {
  int e = blockIdx.x * blockDim.x + threadIdx.x;
  if (e < NE) {
    int dst = (int)ei[NE + e];  // edge_index[1][e]
    atomicAdd(&deg[dst], 1.0f);
  }
}

__global__ void rsqrt_kernel(float* __restrict__ deg) {
  int i = blockIdx.x * blockDim.x + threadIdx.x;
  if (i < NN) {
    float d = deg[i];
    deg[i] = (d > 0.0f) ? rsqrtf(d) : 0.0f;
  }
}

// ---------------- dense GEMM via V_WMMA_F32_16X16X4_F32 ----------------
// H[NN x CH] = X[NN x CH] @ W[CH x CH], row-major.
// One wave per 16x16 output tile; block = 8 waves covering all 8 N-tiles of
// one M-tile. NN = 3125 * 16 exactly, so no remainder handling.

__global__ __launch_bounds__(256) void gemm_wmma_kernel(const float* __restrict__ X,
                                                        const float* __restrict__ W,
                                                        float* __restrict__ H) {
  const int lane  = threadIdx.x & 31;
  const int wave  = threadIdx.x >> 5;
  const int m0    = blockIdx.x * 16;   // row tile
  const int n0    = wave * 16;         // col tile (CH/16 == 8 == waves/block)
  const int l15   = lane & 15;
  const int khalf = (lane >> 4) * 2;   // lanes 0-15: K=0,1 ; lanes 16-31: K=2,3

  v8f c = {0.f, 0.f, 0.f, 0.f, 0.f, 0.f, 0.f, 0.f};
  const float* xrow = X + (size_t)(m0 + l15) * CH;
  const int    n    = n0 + l15;

  #pragma unroll 4
  for (int k0 = 0; k0 < CH; k0 += 4) {
    // A fragment: 16x4 fp32, row m=lane&15, K = k0+khalf, k0+khalf+1 (8B aligned)
    v2f a = *(const v2f*)(xrow + k0 + khalf);
    // B fragment: 4x16 fp32, col n, same K split across lane halves
    v2f b;
    b.x = W[(size_t)(k0 + khalf) * CH + n];
    b.y = W[(size_t)(k0 + khalf + 1) * CH + n];
    c = __builtin_amdgcn_wmma_f32_16x16x4_f32(false, a, false, b, (short)0, c,
                                              false, false);
  }

  // C/D layout: VGPR r -> M = r + (lane>=16 ? 8 : 0), N = lane&15
  const int hi8 = (lane >> 4) * 8;
  #pragma unroll
  for (int r = 0; r < 8; ++r) {
    H[(size_t)(m0 + hi8 + r) * CH + n0 + l15] = c[r];
  }
}

// ---------------- self-loop + bias init: out = h*dinv^2 + b ----------------

__global__ void self_bias_kernel(const float* __restrict__ H,
                                 const float* __restrict__ dinv,
                                 const float* __restrict__ b,
                                 float* __restrict__ out) {
  int i = blockIdx.x * blockDim.x + threadIdx.x;
  if (i < NN * CH) {
    int node = i >> 7;
    int ch   = i & (CH - 1);
    float di = dinv[node];
    out[i] = H[i] * di * di + b[ch];
  }
}

// ---------------- edge scatter: out[dst] += h[src] * dinv[src]*dinv[dst] ----

__global__ __launch_bounds__(256) void scatter_kernel(const long long* __restrict__ ei,
                                                      const float* __restrict__ H,
                                                      const float* __restrict__ dinv,
                                                      float* __restrict__ out) {
  int t    = blockIdx.x * blockDim.x + threadIdx.x;
  int e    = t >> 5;         // one wave32 per edge
  int lane = t & 31;         // each lane: 4 contiguous channels
  if (e < NE) {
    int src = (int)ei[e];
    int dst = (int)ei[NE + e];
    float norm = dinv[src] * dinv[dst];
    const float4 v = *(const float4*)(H + (size_t)src * CH + lane * 4);
    float* o = out + (size_t)dst * CH + lane * 4;
    atomicAdd(o + 0, v.x * norm);
    atomicAdd(o + 1, v.y * norm);
    atomicAdd(o + 2, v.z * norm);
    atomicAdd(o + 3, v.w * norm);
  }
}

__global__ void relu_kernel(float* __restrict__ x) {
  int i = blockIdx.x * blockDim.x + threadIdx.x;
  if (i < NN * CH) x[i] = fmaxf(x[i], 0.0f);
}

// ---------------- launch ----------------

extern "C" void kernel_launch(void* const* d_in, const int* in_sizes, int n_in,
                              void* d_out, int out_size, void* d_ws, size_t ws_size,
                              hipStream_t stream) {
  const float*     x  = (const float*)d_in[0];
  const long long* ei = (const long long*)d_in[1];  // int64 per reference
  const float*     W1 = (const float*)d_in[2];
  const float*     b1 = (const float*)d_in[3];
  const float*     W2 = (const float*)d_in[4];
  const float*     b2 = (const float*)d_in[5];
  float*           out = (float*)d_out;

  // workspace layout: dinv (50000 f, padded to 256KB) | bufA | bufB
  char*  ws   = (char*)d_ws;
  float* dinv = (float*)ws;
  float* bufA = (float*)(ws + (256 << 10));
  float* bufB = (float*)(ws + (256 << 10) + (size_t)NN * CH * sizeof(float));

  const int B = 256;
  const int gN  = (NN + B - 1) / B;              // node-count grids
  const int gE  = (NE + B - 1) / B;              // edge-count grids
  const int gNC = (NN * CH + B - 1) / B;         // node*channel grids
  const int gEW = (NE * 32 + B - 1) / B;         // wave-per-edge grids
  const int gM  = NN / 16;                       // 3125 GEMM M-tiles

  // normalization
  init_deg_kernel<<<gN, B, 0, stream>>>(dinv);
  deg_acc_kernel<<<gE, B, 0, stream>>>(ei, dinv);
  rsqrt_kernel<<<gN, B, 0, stream>>>(dinv);

  // layer 1: h1 = x@W1 ; agg1 = scatter(h1)+b1 ; relu
  gemm_wmma_kernel<<<gM, B, 0, stream>>>(x, W1, bufA);
  self_bias_kernel<<<gNC, B, 0, stream>>>(bufA, dinv, b1, bufB);
  scatter_kernel<<<gEW, B, 0, stream>>>(ei, bufA, dinv, bufB);
  relu_kernel<<<gNC, B, 0, stream>>>(bufB);

  // layer 2: h2 = agg1@W2 ; out = scatter(h2)+b2
  gemm_wmma_kernel<<<gM, B, 0, stream>>>(bufB, W2, bufA);
  self_bias_kernel<<<gNC, B, 0, stream>>>(bufA, dinv, b2, out);
  scatter_kernel<<<gEW, B, 0, stream>>>(ei, bufA, dinv, out);
}